// compressor_17454747091102
// MI455X (gfx1250) — compile-verified
//
#include <hip/hip_runtime.h>
#include <hip/hip_bf16.h>

// Bitwise majority-of-8 vote over packed sign words.
//   out[m] bit i = 1  iff  >4 of the 8 voters have bit i set in word m.
// Memory-bound streaming kernel: input staged via CDNA5 async global->LDS DMA
// (global_load_async_to_lds_b128 th:TH_LOAD_NT / s_wait_asynccnt),
// double-buffered per WG. All global traffic is single-use -> non-temporal.

typedef unsigned uv4 __attribute__((ext_vector_type(4)));

#define NVOTER 8
#define TPB 256
#define WPT 4                      // words per thread per tile (one B128)
#define TILE_WORDS (TPB * WPT)     // 1024 words per tile

// Majority of 8 one-bit inputs, computed bitwise over 32 lanes of a word.
// Full-adder network: count = S0 + 2*T, T = sE + 2*(cD + cE).
// count >= 5  <=>  (cD & cE) | ((cD ^ cE) & (sE | S0))
template <typename T>
__device__ __forceinline__ T maj8(T a0, T a1, T a2, T a3, T a4, T a5, T a6, T a7) {
    T t1 = a0 ^ a1, sA = t1 ^ a2, cA = (a0 & a1) | (t1 & a2);
    T t2 = a3 ^ a4, sB = t2 ^ a5, cB = (a3 & a4) | (t2 & a5);
    T sC = a6 ^ a7, cC = a6 & a7;
    T t3 = sA ^ sB, S0 = t3 ^ sC, C1 = (sA & sB) | (t3 & sC);
    T t4 = cA ^ cB, sD = t4 ^ cC, cD = (cA & cB) | (t4 & cC);
    T sE = sD ^ C1, cE = sD & C1;
    T u  = cD ^ cE, b  = cD & cE;
    return b | (u & (sE | S0));
}

__global__ __launch_bounds__(TPB) void majority8_kernel(
        const unsigned* __restrict__ src,  // NVOTER rows x M words, row stride M
        unsigned* __restrict__ out,        // M words
        int M, int nfull) {
    // Double-buffered staging: each thread owns 8 x 16B slots per buffer and
    // only ever reads back its own DMA'd data -> no workgroup barriers needed.
    // WAR on a buffer is covered by the compiler's s_wait_dscnt before the
    // consumed values feed the VALU network.
    __shared__ uv4 stage[2][NVOTER][TPB];

    const int tid = threadIdx.x;

    auto prefetch = [&](int tile, int buf) {
        const unsigned w = (unsigned)tile * TILE_WORDS + (unsigned)tid * WPT;
#pragma unroll
        for (int v = 0; v < NVOTER; ++v) {
            unsigned long long ga =
                (unsigned long long)(src + (size_t)v * (size_t)M + w);
            unsigned la = (unsigned)(unsigned long long)&stage[buf][v][tid];
            // CDNA5 async DMA: global -> LDS, 16 bytes per lane, ASYNCcnt++.
            // Data is single-use: non-temporal load policy.
            asm volatile("global_load_async_to_lds_b128 %0, %1, off th:TH_LOAD_NT"
                         :: "v"(la), "v"(ga) : "memory");
        }
    };

    int i = blockIdx.x;
    int buf = 0;
    if (i < nfull) prefetch(i, 0);

    while (i < nfull) {
        const int nxt = i + gridDim.x;
        const bool more = nxt < nfull;
        if (more) prefetch(nxt, buf ^ 1);          // keep next tile in flight

        // Async loads complete in order: <=8 outstanding means tile i landed.
        if (more) asm volatile("s_wait_asynccnt 0x8" ::: "memory");
        else      asm volatile("s_wait_asynccnt 0x0" ::: "memory");

        uv4 r0 = stage[buf][0][tid];
        uv4 r1 = stage[buf][1][tid];
        uv4 r2 = stage[buf][2][tid];
        uv4 r3 = stage[buf][3][tid];
        uv4 r4 = stage[buf][4][tid];
        uv4 r5 = stage[buf][5][tid];
        uv4 r6 = stage[buf][6][tid];
        uv4 r7 = stage[buf][7][tid];

        uv4 m = maj8(r0, r1, r2, r3, r4, r5, r6, r7);

        const unsigned w = (unsigned)i * TILE_WORDS + (unsigned)tid * WPT;
        __builtin_nontemporal_store(m, (uv4*)(out + w));   // streamed once

        buf ^= 1;
        i = nxt;
    }

    // Scalar tail: covers M not a multiple of TILE_WORDS (or misaligned M).
    const long long base = (long long)nfull * TILE_WORDS;
    for (long long w = base + (long long)blockIdx.x * blockDim.x + tid; w < M;
         w += (long long)gridDim.x * blockDim.x) {
        unsigned a0 = __builtin_nontemporal_load(src + 0 * (size_t)M + w);
        unsigned a1 = __builtin_nontemporal_load(src + 1 * (size_t)M + w);
        unsigned a2 = __builtin_nontemporal_load(src + 2 * (size_t)M + w);
        unsigned a3 = __builtin_nontemporal_load(src + 3 * (size_t)M + w);
        unsigned a4 = __builtin_nontemporal_load(src + 4 * (size_t)M + w);
        unsigned a5 = __builtin_nontemporal_load(src + 5 * (size_t)M + w);
        unsigned a6 = __builtin_nontemporal_load(src + 6 * (size_t)M + w);
        unsigned a7 = __builtin_nontemporal_load(src + 7 * (size_t)M + w);
        out[w] = maj8(a0, a1, a2, a3, a4, a5, a6, a7);
    }
}

extern "C" void kernel_launch(void* const* d_in, const int* in_sizes, int n_in,
                              void* d_out, int out_size, void* d_ws, size_t ws_size,
                              hipStream_t stream) {
    // d_in[0] = para_temp (unused float), d_in[1] = packed sign words (8 x M int32)
    const unsigned* src = (const unsigned*)d_in[1];
    unsigned* out = (unsigned*)d_out;   // int32 bit patterns

    const int total = in_sizes[1];
    const int M = total / NVOTER;       // M = 2^20 for the harness shapes

    // Async B128 path needs 16B-aligned voter rows: require M % 4 == 0.
    int nfull = ((M % 4) == 0) ? (M / TILE_WORDS) : 0;

    int grid = nfull;
    if (grid > 512) grid = 512;         // 512 WGs => 2 tiles/WG: overlap DMA+compute
    if (grid < 1) {
        grid = (M + TPB - 1) / TPB;
        if (grid > 512) grid = 512;
        if (grid < 1) grid = 1;
    }

    majority8_kernel<<<dim3(grid), dim3(TPB), 0, stream>>>(src, out, M, nfull);
}